// DynamicGraphBinaryClassificationModel_28509992910998
// MI455X (gfx1250) — compile-verified
//
#include <hip/hip_runtime.h>
#include <hip/hip_bf16.h>
#include <math.h>

// ---------------------------------------------------------------------------
// GCN forward for MI455X (gfx1250). fp32 everywhere (matches reference).
// Dense transforms: V_WMMA_F32_16X16X4_F32 (CDNA5 wave32 WMMA).
// Weight staging:   TENSOR_LOAD_TO_LDS (TDM async DMA) + s_wait_tensorcnt.
// ---------------------------------------------------------------------------

#define FEAT 128

typedef __attribute__((ext_vector_type(2))) float        v2f;
typedef __attribute__((ext_vector_type(8))) float        v8f;
typedef __attribute__((ext_vector_type(4))) unsigned int v4u;
typedef __attribute__((ext_vector_type(8))) int          v8i;
typedef __attribute__((ext_vector_type(4))) int          v4i;

// ---------------- degree / normalization ----------------------------------

__global__ void k_deg_init(float* __restrict__ deg, int n) {
    int i = blockIdx.x * blockDim.x + threadIdx.x;
    if (i < n) deg[i] = 1.0f;  // self loop
}

__global__ void k_deg_count(const long long* __restrict__ ei, float* __restrict__ deg, int nE) {
    int e = blockIdx.x * blockDim.x + threadIdx.x;
    if (e < nE) {
        long long d = ei[(size_t)nE + e];  // dst row
        atomicAdd(&deg[d], 1.0f);
    }
}

__global__ void k_deg_finalize(float* __restrict__ deg, int n) {
    int i = blockIdx.x * blockDim.x + threadIdx.x;
    if (i < n) deg[i] = rsqrtf(fmaxf(deg[i], 1.0f));  // now holds dinv
}

// ---------------- TDM: DMA a 128x128 f32 tensor (64 KB) into LDS -----------
// 2D descriptor, tile == tensor, data_size = 4 bytes. Groups 2/3 zero
// (tile_dim2 == 0 disables higher dims). Issued by a single wave; that wave
// waits on TENSORcnt, then the workgroup barrier publishes LDS.

#if __has_builtin(__builtin_amdgcn_tensor_load_to_lds)
#define HAVE_TDM 1
__device__ __forceinline__ void tdm_load_128x128_f32(const float* gsrc, unsigned lds_off) {
    unsigned long long ga = (unsigned long long)(uintptr_t)gsrc;
    v4u g0;
    g0.x = 0x1u;                                              // count=1 (valid user D#)
    g0.y = lds_off;                                           // lds_addr (bytes)
    g0.z = (unsigned)(ga & 0xFFFFFFFFu);                      // global_addr[31:0]
    g0.w = (unsigned)((ga >> 32) & 0x01FFFFFFu) | (2u << 30); // addr[56:32] | type=2
    v8i g1;
    g1[0] = (int)(2u << 16);    // workgroup_mask=0, data_size=2 (4 bytes)
    g1[1] = (int)(128u << 16);  // tensor_dim0[15:0] @ bits[63:48] = 128
    g1[2] = (int)(128u << 16);  // tensor_dim0 hi=0; tensor_dim1[15:0] @ bits[95:80] = 128
    g1[3] = (int)(128u << 16);  // tensor_dim1 hi=0; tile_dim0 @ bits[127:112] = 128
    g1[4] = (int)(128u);        // tile_dim1 @ bits[143:128] = 128; tile_dim2 = 0
    g1[5] = (int)(128u);        // tensor_dim0_stride[31:0] = 128 elements
    g1[6] = 0;                  // stride hi / dim1_stride lo
    g1[7] = 0;
    v4i gz = {0, 0, 0, 0};
#if defined(__clang_major__) && (__clang_major__ >= 23)
    v8i gz8 = {0, 0, 0, 0, 0, 0, 0, 0};
    __builtin_amdgcn_tensor_load_to_lds(g0, g1, gz, gz, gz8, 0);
#else
    __builtin_amdgcn_tensor_load_to_lds(g0, g1, gz, gz, 0);
#endif
    __builtin_amdgcn_s_wait_tensorcnt(0);
}
#else
#define HAVE_TDM 0
#endif

// ---------------- WMMA GEMM: C[n,128] = A[n,128] @ W[128,128] --------------
// 256 threads = 8 waves. Block covers 128 rows; wave w owns rows
// [blk*128 + 16w, +16). W staged in LDS (64 KB) by the TDM. Wave computes
// 8 n-tiles of 16x16 via v_wmma_f32_16x16x4_f32, K-loop of 32 steps.

__global__ __launch_bounds__(256) void k_gemm_wmma(
    const float* __restrict__ A, const float* __restrict__ W,
    float* __restrict__ C, int nrows)
{
    __shared__ float Ws[FEAT * FEAT];

#if HAVE_TDM
    if (threadIdx.x < 32) {
        tdm_load_128x128_f32(W, (unsigned)(uintptr_t)(void*)Ws);
    }
    __syncthreads();
#else
    for (int i = threadIdx.x; i < FEAT * FEAT; i += 256) Ws[i] = W[i];
    __syncthreads();
#endif

    const int wave  = threadIdx.x >> 5;   // 0..7
    const int lane  = threadIdx.x & 31;
    const int lm    = lane & 15;
    const int khalf = lane >> 4;          // 0: K{0,1}, 1: K{2,3}
    const int m0    = blockIdx.x * 128 + wave * 16;
    const int row   = m0 + lm;
    const bool rowOK = row < nrows;

    v8f acc[8] = {};

    for (int k = 0; k < FEAT; k += 4) {
        // A fragment (16x4 f32): lane holds A[row][k + 2*khalf + {0,1}]
        v2f a; a.x = 0.0f; a.y = 0.0f;
        if (rowOK) {
            const float* ap = A + (size_t)row * FEAT + k + khalf * 2;
            a.x = ap[0];
            a.y = ap[1];
        }
#pragma unroll
        for (int t = 0; t < 8; ++t) {
            const int n = t * 16 + lm;
            // B fragment (4x16 f32): lane holds W[k + 2*khalf + {0,1}][n]
            v2f b;
            b.x = Ws[(k + khalf * 2 + 0) * FEAT + n];
            b.y = Ws[(k + khalf * 2 + 1) * FEAT + n];
            acc[t] = __builtin_amdgcn_wmma_f32_16x16x4_f32(
                false, a, false, b, (short)0, acc[t], false, false);
        }
    }

    // C/D layout: VGPR v, lanes 0-15 -> M = v, lanes 16-31 -> M = v + 8
    const int mbase = m0 + khalf * 8;
#pragma unroll
    for (int t = 0; t < 8; ++t) {
        const int n = t * 16 + lm;
#pragma unroll
        for (int v = 0; v < 8; ++v) {
            const int r = mbase + v;
            if (r < nrows) C[(size_t)r * FEAT + n] = acc[t][v];
        }
    }
}

// ---------------- aggregation: self-loop init + edge scatter ---------------

// out[i][c] = bias[c] + hT[i][c] * dinv[i]^2
__global__ void k_agg_init(const float* __restrict__ hT, const float* __restrict__ bias,
                           const float* __restrict__ dinv, float* __restrict__ out, int n)
{
    int idx = blockIdx.x * blockDim.x + threadIdx.x;
    if (idx >= n * FEAT) return;
    int r = idx >> 7, c = idx & (FEAT - 1);
    float w = dinv[r];
    out[idx] = bias[c] + hT[idx] * (w * w);
}

// one wave per edge: out[dst][c] += hT[src][c] * dinv[src]*dinv[dst]
__global__ __launch_bounds__(256) void k_edge_scatter(
    const long long* __restrict__ ei, const float* __restrict__ hT,
    const float* __restrict__ dinv, float* __restrict__ out, int nE)
{
    int e    = blockIdx.x * 8 + (threadIdx.x >> 5);
    int lane = threadIdx.x & 31;
    if (e >= nE) return;
    long long s = ei[e];
    long long d = ei[(size_t)nE + e];
    float w = dinv[s] * dinv[d];
    const float4 v = *(const float4*)(hT + (size_t)s * FEAT + lane * 4);
    float* op = out + (size_t)d * FEAT + lane * 4;
    atomicAdd(op + 0, v.x * w);
    atomicAdd(op + 1, v.y * w);
    atomicAdd(op + 2, v.z * w);
    atomicAdd(op + 3, v.w * w);
}

// ---------------- batch norm (training-mode stats) + ReLU ------------------

__global__ void k_bn_zero(float* __restrict__ bnsum, float* __restrict__ bnsq) {
    int c = threadIdx.x;
    if (c < FEAT) { bnsum[c] = 0.0f; bnsq[c] = 0.0f; }
}

__global__ __launch_bounds__(128) void k_bn_stats(
    const float* __restrict__ h, float* __restrict__ bnsum, float* __restrict__ bnsq, int n)
{
    int c = threadIdx.x;  // 128 threads = 1 channel each
    float s = 0.0f, q = 0.0f;
    for (int r = blockIdx.x; r < n; r += gridDim.x) {
        float v = h[(size_t)r * FEAT + c];
        s += v; q += v * v;
    }
    atomicAdd(&bnsum[c], s);
    atomicAdd(&bnsq[c], q);
}

__global__ void k_bn_finalize(const float* __restrict__ bnsum, const float* __restrict__ bnsq,
                              const float* __restrict__ g, const float* __restrict__ be,
                              float* __restrict__ scaleC, float* __restrict__ shiftC, int n)
{
    int c = threadIdx.x;
    if (c >= FEAT) return;
    float inv_n = 1.0f / (float)n;
    float mu   = bnsum[c] * inv_n;
    float var  = bnsq[c] * inv_n - mu * mu;
    float rstd = rsqrtf(var + 1e-5f);
    float sc   = g[c] * rstd;
    scaleC[c] = sc;
    shiftC[c] = be[c] - mu * sc;
}

__global__ void k_bn_apply_relu(float* __restrict__ h, const float* __restrict__ scaleC,
                                const float* __restrict__ shiftC, int total)
{
    int idx = blockIdx.x * blockDim.x + threadIdx.x;
    if (idx >= total) return;
    int c = idx & (FEAT - 1);
    float v = h[idx] * scaleC[c] + shiftC[c];
    h[idx] = fmaxf(v, 0.0f);
}

// ---------------- mean pooling + sigmoid head ------------------------------

__global__ void k_pool_zero(float* __restrict__ pooled, float* __restrict__ counts, int g) {
    int i = blockIdx.x * blockDim.x + threadIdx.x;
    if (i < g * FEAT) pooled[i] = 0.0f;
    if (i < g) counts[i] = 0.0f;
}

__global__ __launch_bounds__(128) void k_pool_accum(
    const float* __restrict__ h, const long long* __restrict__ batch,
    float* __restrict__ pooled, float* __restrict__ counts, int n)
{
    int c = threadIdx.x;
    for (int r = blockIdx.x; r < n; r += gridDim.x) {
        int g = (int)batch[r];
        atomicAdd(&pooled[(size_t)g * FEAT + c], h[(size_t)r * FEAT + c]);
        if (c == 0) atomicAdd(&counts[g], 1.0f);
    }
}

__global__ __launch_bounds__(128) void k_head(
    const float* __restrict__ pooled, const float* __restrict__ counts,
    const float* __restrict__ Wout, const float* __restrict__ bout,
    float* __restrict__ out)
{
    __shared__ float red[FEAT];
    int g = blockIdx.x, t = threadIdx.x;
    red[t] = pooled[(size_t)g * FEAT + t] * Wout[t];
    __syncthreads();
    for (int s = 64; s > 0; s >>= 1) {
        if (t < s) red[t] += red[t + s];
        __syncthreads();
    }
    if (t == 0) {
        float cnt = fmaxf(counts[g], 1.0f);
        float z = red[0] / cnt + bout[0];
        out[g] = 1.0f / (1.0f + expf(-z));
    }
}

// ---------------------------------------------------------------------------

extern "C" void kernel_launch(void* const* d_in, const int* in_sizes, int n_in,
                              void* d_out, int out_size, void* d_ws, size_t ws_size,
                              hipStream_t stream) {
    (void)n_in; (void)ws_size;

    const float*     x     = (const float*)d_in[0];
    const long long* ei    = (const long long*)d_in[1];
    const long long* batch = (const long long*)d_in[2];
    const float* W[3]  = { (const float*)d_in[3],  (const float*)d_in[7],  (const float*)d_in[11] };
    const float* b[3]  = { (const float*)d_in[4],  (const float*)d_in[8],  (const float*)d_in[12] };
    const float* gm[3] = { (const float*)d_in[5],  (const float*)d_in[9],  (const float*)d_in[13] };
    const float* be[3] = { (const float*)d_in[6],  (const float*)d_in[10], (const float*)d_in[14] };
    const float* Wout  = (const float*)d_in[15];
    const float* bout  = (const float*)d_in[16];
    float* out = (float*)d_out;

    const int N = in_sizes[0] / FEAT;   // 50000
    const int E = in_sizes[1] / 2;      // 800000
    const int G = out_size;             // 64

    // workspace carve-up (floats)
    float* ws     = (float*)d_ws;
    float* dinv   = ws;                       // N
    float* hA     = dinv + N;                 // N*128 (GEMM out)
    float* hB     = hA + (size_t)N * FEAT;    // N*128 (aggregated / layer out)
    float* bnsum  = hB + (size_t)N * FEAT;    // 128
    float* bnsq   = bnsum + FEAT;             // 128
    float* scaleC = bnsq + FEAT;              // 128
    float* shiftC = scaleC + FEAT;            // 128
    float* pooled = shiftC + FEAT;            // G*128
    float* counts = pooled + (size_t)G * FEAT;// G

    const int TOT = N * FEAT;

    // degree normalization
    k_deg_init<<<(N + 255) / 256, 256, 0, stream>>>(dinv, N);
    k_deg_count<<<(E + 255) / 256, 256, 0, stream>>>(ei, dinv, E);
    k_deg_finalize<<<(N + 255) / 256, 256, 0, stream>>>(dinv, N);

    const int gemmGrid = (N + 127) / 128;
    const float* in = x;
    for (int l = 0; l < 3; ++l) {
        k_gemm_wmma<<<gemmGrid, 256, 0, stream>>>(in, W[l], hA, N);
        k_agg_init<<<(TOT + 255) / 256, 256, 0, stream>>>(hA, b[l], dinv, hB, N);
        k_edge_scatter<<<(E + 7) / 8, 256, 0, stream>>>(ei, hA, dinv, hB, E);
        k_bn_zero<<<1, 128, 0, stream>>>(bnsum, bnsq);
        k_bn_stats<<<512, 128, 0, stream>>>(hB, bnsum, bnsq, N);
        k_bn_finalize<<<1, 128, 0, stream>>>(bnsum, bnsq, gm[l], be[l], scaleC, shiftC, N);
        k_bn_apply_relu<<<(TOT + 255) / 256, 256, 0, stream>>>(hB, scaleC, shiftC, TOT);
        in = hB;
    }

    // readout
    k_pool_zero<<<(G * FEAT + 255) / 256, 256, 0, stream>>>(pooled, counts, G);
    k_pool_accum<<<512, 128, 0, stream>>>(hB, batch, pooled, counts, N);
    k_head<<<G, 128, 0, stream>>>(pooled, counts, Wout, bout, out);
}